// L1AttnF_78383153152692
// MI455X (gfx1250) — compile-verified
//
#include <hip/hip_runtime.h>

// L1 attention forward: c[b,h,s,t] = -(1/8) * sum_w |q[b,t,h,w] - k[b,s,h,w]|
// bs=2, n_ctx=512, n_heads=8, width=64 (fixed by harness)

#define NCTX 512
#define NH   8
#define WID  64
#define TT   128   // query-rows (t) per workgroup
#define TS   64    // key-rows   (s) per workgroup
#define NTHR 256

typedef float v4f __attribute__((ext_vector_type(4)));
typedef int   v4i __attribute__((ext_vector_type(4)));

#if defined(__HIP_DEVICE_COMPILE__) && __has_builtin(__builtin_amdgcn_global_load_async_to_lds_b128)
#define USE_ASYNC 1
#else
#define USE_ASYNC 0
#endif

__device__ __forceinline__ void cp16(const float* g, float* l) {
#if USE_ASYNC
    // gfx1250 async direct global->LDS copy, 16B per lane, tracked by ASYNCcnt.
    // Builtin expects v4i* params (per clang diagnostic); C-style cast drops const.
    __builtin_amdgcn_global_load_async_to_lds_b128((v4i*)g, (v4i*)l, 0, 0);
#else
    *(v4f*)l = *(const v4f*)g;
#endif
}

__device__ __forceinline__ void wait_async() {
#if USE_ASYNC
#if __has_builtin(__builtin_amdgcn_s_wait_asynccnt)
    __builtin_amdgcn_s_wait_asynccnt(0);
#else
    asm volatile("s_wait_asynccnt 0" ::: "memory");
#endif
#endif
}

// Swizzled [w][row] compute-layout indices (XOR swizzle keeps LDS banks spread
// for both the transpose scatter-stores and the b128 compute loads).
__device__ __forceinline__ int qidx(int w, int t) {
    int cb = (t >> 3) ^ ((w >> 2) & 15);      // 16 blocks of 8 floats
    return w * TT + cb * 8 + (t & 7);
}
__device__ __forceinline__ int kidx(int w, int s) {
    int cb = (s >> 2) ^ ((w >> 2) & 15);      // 16 blocks of 4 floats
    return w * TS + cb * 4 + (s & 3);
}

__global__ __launch_bounds__(NTHR) void l1attn_fwd_kernel(
        const float* __restrict__ q, const float* __restrict__ k,
        float* __restrict__ out) {
    __shared__ float sm[TT * WID + TS * WID];     // 48 KB
    float* Qc = sm;                               // 8192 dwords
    float* Kc = sm + TT * WID;                    // 4096 dwords

    const int tid = threadIdx.x;
    const int t0  = blockIdx.x * TT;
    const int s0  = blockIdx.y * TS;
    const int bh  = blockIdx.z;                   // b*NH + h
    const int b   = bh >> 3;
    const int h   = bh & 7;

    const int rstride = NH * WID;                 // 512 floats between rows
    const size_t gbase = ((size_t)b * NCTX * NH + h) * WID;

    // ---- Stage tiles (natural [row][w] layout) straight into LDS ----
    // Q: 2048 x 16B chunks (8/thread), K: 1024 (4/thread); lanes -> consecutive
    // chunks => coalesced 16B/lane global traffic.
#pragma unroll
    for (int i = 0; i < 8; ++i) {
        int c = i * NTHR + tid;
        int t = c >> 4;
        int w = (c & 15) * 4;
        cp16(q + gbase + (size_t)(t0 + t) * rstride + w, Qc + c * 4);
    }
#pragma unroll
    for (int i = 0; i < 4; ++i) {
        int c = i * NTHR + tid;
        int s = c >> 4;
        int w = (c & 15) * 4;
        cp16(k + gbase + (size_t)(s0 + s) * rstride + w, Kc + c * 4);
    }
    wait_async();
    __syncthreads();

    // ---- In-place transpose to swizzled [w][row] layout ----
    // Read everything this thread owns to VGPRs, barrier, scatter-store.
    v4f rq[8], rk[4];
    int tq[8], wq[8], tk[4], wk[4];
#pragma unroll
    for (int i = 0; i < 8; ++i) {
        int c = i * NTHR + tid;
        tq[i] = c >> 4; wq[i] = (c & 15) * 4;
        rq[i] = *(const v4f*)(Qc + c * 4);
    }
#pragma unroll
    for (int i = 0; i < 4; ++i) {
        int c = i * NTHR + tid;
        tk[i] = c >> 4; wk[i] = (c & 15) * 4;
        rk[i] = *(const v4f*)(Kc + c * 4);
    }
    asm volatile("s_wait_dscnt 0" ::: "memory");  // reads done before anyone overwrites
    __syncthreads();
#pragma unroll
    for (int i = 0; i < 8; ++i)
#pragma unroll
        for (int e = 0; e < 4; ++e)
            Qc[qidx(wq[i] + e, tq[i])] = rq[i][e];
#pragma unroll
    for (int i = 0; i < 4; ++i)
#pragma unroll
        for (int e = 0; e < 4; ++e)
            Kc[kidx(wk[i] + e, tk[i])] = rk[i][e];
    __syncthreads();

    // ---- Compute: 8(t) x 4(s) register block per thread ----
    const int tx = tid & 15;   // t block: tx*8 .. tx*8+7
    const int ty = tid >> 4;   // s block: ty*4 .. ty*4+3
    float acc[8][4] = {};

#pragma unroll 4
    for (int w = 0; w < WID; ++w) {
        const int sw = (w >> 2) & 15;
        const int qoff = w * TT + (tx ^ sw) * 8;
        v4f qa = *(const v4f*)(Qc + qoff);
        v4f qb = *(const v4f*)(Qc + qoff + 4);
        v4f kv = *(const v4f*)(Kc + w * TS + (ty ^ sw) * 4);
        float qv[8] = {qa.x, qa.y, qa.z, qa.w, qb.x, qb.y, qb.z, qb.w};
        float kw[4] = {kv.x, kv.y, kv.z, kv.w};
#pragma unroll
        for (int i = 0; i < 8; ++i)
#pragma unroll
            for (int j = 0; j < 4; ++j)
                acc[i][j] += __builtin_fabsf(qv[i] - kw[j]);  // v_sub + v_add with |src|
    }

    // ---- Store: out[bh][s][t], t contiguous -> coalesced b128 stores ----
    const float scale = -0.125f;  // -1/sqrt(64)
    const size_t obase = (size_t)bh * NCTX * NCTX;
#pragma unroll
    for (int j = 0; j < 4; ++j) {
        size_t row = obase + (size_t)(s0 + ty * 4 + j) * NCTX + t0 + tx * 8;
        v4f o0 = {scale * acc[0][j], scale * acc[1][j], scale * acc[2][j], scale * acc[3][j]};
        v4f o1 = {scale * acc[4][j], scale * acc[5][j], scale * acc[6][j], scale * acc[7][j]};
        *(v4f*)(out + row)     = o0;
        *(v4f*)(out + row + 4) = o1;
    }
}

extern "C" void kernel_launch(void* const* d_in, const int* in_sizes, int n_in,
                              void* d_out, int out_size, void* d_ws, size_t ws_size,
                              hipStream_t stream) {
    (void)in_sizes; (void)n_in; (void)out_size; (void)d_ws; (void)ws_size;
    const float* q = (const float*)d_in[0];
    const float* k = (const float*)d_in[1];
    float* out = (float*)d_out;
    dim3 grid(NCTX / TT, NCTX / TS, 2 * NH);  // (4, 8, 16) = 512 WGs
    l1attn_fwd_kernel<<<grid, dim3(NTHR), 0, stream>>>(q, k, out);
}